// DepthLossWithMask_50663434224037
// MI455X (gfx1250) — compile-verified
//
#include <hip/hip_runtime.h>
#include <cstdint>

#define BLOCK 256
#define NWAVES (BLOCK / 32)

// ---- CDNA5 async global->LDS copy (ASYNCcnt-tracked), inline asm ----
__device__ __forceinline__ void async_copy_b128(uint32_t lds_addr, uint64_t gaddr) {
    // copies 16 bytes from each lane's global address in gaddr to its LDS address in lds_addr
    asm volatile("global_load_async_to_lds_b128 %0, %1, off"
                 :: "v"(lds_addr), "v"(gaddr)
                 : "memory");
}
__device__ __forceinline__ void s_wait_async_le3() {
    asm volatile("s_wait_asynccnt 0x3" ::: "memory");
}
__device__ __forceinline__ void s_wait_async_0() {
    asm volatile("s_wait_asynccnt 0x0" ::: "memory");
}

// ---- Kernel 1: streaming masked-L1 partial reduction, async double-buffered ----
__global__ __launch_bounds__(BLOCK) void masked_l1_partial(
    const float* __restrict__ outp, const float* __restrict__ lab0,
    const float* __restrict__ lab1, int n, int n4,
    float* __restrict__ loss_part, unsigned* __restrict__ cnt_part) {

    __shared__ __align__(16) float4 stage[2][3][BLOCK];  // 24 KB double buffer
    __shared__ float    wsum[NWAVES];
    __shared__ unsigned wcnt[NWAVES];

    const int tid = threadIdx.x;
    const uint32_t lds_base = (uint32_t)(uintptr_t)&stage[0][0][0];

    const float4* o4 = (const float4*)outp;
    const float4* a4 = (const float4*)lab0;
    const float4* w4 = (const float4*)lab1;

    const int stride = (int)gridDim.x * BLOCK;            // uniform
    const int base   = (int)blockIdx.x * BLOCK + tid;     // per-lane
    const int K      = (n4 > 0) ? (n4 + stride - 1) / stride : 0;  // uniform trip count

    float    loss = 0.0f;
    unsigned cnt  = 0u;

    auto issue = [&](int buf, int idx) {
        int c = (idx < n4) ? idx : (n4 > 0 ? n4 - 1 : 0);   // clamp: keep EXEC full
        uint32_t l = lds_base + (uint32_t)((buf * 3 * BLOCK + tid) << 4);
        async_copy_b128(l,                      (uint64_t)(uintptr_t)(o4 + c));
        async_copy_b128(l + (BLOCK << 4),       (uint64_t)(uintptr_t)(a4 + c));
        async_copy_b128(l + 2u * (BLOCK << 4),  (uint64_t)(uintptr_t)(w4 + c));
    };

    if (K > 0) {
        int idx = base;
        int buf = 0;
        issue(0, idx);                          // prologue: stage 0 in flight
        for (int k = 0; k < K; ++k) {
            int nidx = idx + stride;
            if (k + 1 < K) {                    // uniform branch
                issue(buf ^ 1, nidx);           // 3 more asyncs in flight
                s_wait_async_le3();             // current stage's 3 done
            } else {
                s_wait_async_0();               // drain
            }
            float4 o = stage[buf][0][tid];      // ds_load_b128 x3
            float4 a = stage[buf][1][tid];
            float4 w = stage[buf][2][tid];
            if (idx < n4) {
                loss += fabsf(o.x - a.x) * w.x;
                loss += fabsf(o.y - a.y) * w.y;
                loss += fabsf(o.z - a.z) * w.z;
                loss += fabsf(o.w - a.w) * w.w;
                cnt += (unsigned)((a.x != 0.0f) + (a.y != 0.0f) +
                                  (a.z != 0.0f) + (a.w != 0.0f));
            }
            buf ^= 1;
            idx = nidx;
        }
    }

    // Scalar tail (n % 4 != 0), handled by block 0 only.
    if (blockIdx.x == 0) {
        for (int i = n4 * 4 + tid; i < n; i += BLOCK) {
            float a = lab0[i];
            loss += fabsf(outp[i] - a) * lab1[i];
            cnt  += (unsigned)(a != 0.0f);
        }
    }

    // wave32 reduction
    for (int off = 16; off > 0; off >>= 1) {
        loss += __shfl_xor(loss, off, 32);
        cnt  += __shfl_xor(cnt,  off, 32);
    }
    const int lane = tid & 31, wid = tid >> 5;
    if (lane == 0) { wsum[wid] = loss; wcnt[wid] = cnt; }
    __syncthreads();
    if (wid == 0) {
        float    l = (lane < NWAVES) ? wsum[lane] : 0.0f;
        unsigned c = (lane < NWAVES) ? wcnt[lane] : 0u;
        for (int off = NWAVES / 2; off > 0; off >>= 1) {
            l += __shfl_xor(l, off, 32);
            c += __shfl_xor(c, off, 32);
        }
        if (lane == 0) { loss_part[blockIdx.x] = l; cnt_part[blockIdx.x] = c; }
    }
}

// ---- Kernel 2: deterministic finalize (single block) ----
__global__ __launch_bounds__(BLOCK) void masked_l1_final(
    const float* __restrict__ loss_part, const unsigned* __restrict__ cnt_part,
    int G, float* __restrict__ out) {

    __shared__ float    wsum[NWAVES];
    __shared__ unsigned wcnt[NWAVES];

    const int tid = threadIdx.x;
    float    l = 0.0f;
    unsigned c = 0u;
    for (int i = tid; i < G; i += BLOCK) { l += loss_part[i]; c += cnt_part[i]; }
    for (int off = 16; off > 0; off >>= 1) {
        l += __shfl_xor(l, off, 32);
        c += __shfl_xor(c, off, 32);
    }
    const int lane = tid & 31, wid = tid >> 5;
    if (lane == 0) { wsum[wid] = l; wcnt[wid] = c; }
    __syncthreads();
    if (tid == 0) {
        float    L = 0.0f;
        unsigned C = 0u;
        for (int i = 0; i < NWAVES; ++i) { L += wsum[i]; C += wcnt[i]; }
        out[0] = (C == 0u) ? 0.0f : L / (float)C;
    }
}

extern "C" void kernel_launch(void* const* d_in, const int* in_sizes, int n_in,
                              void* d_out, int out_size, void* d_ws, size_t ws_size,
                              hipStream_t stream) {
    const float* outp = (const float*)d_in[0];
    const float* lab0 = (const float*)d_in[1];
    const float* lab1 = (const float*)d_in[2];
    const int n  = in_sizes[0];
    const int n4 = n / 4;

    // ~8 float4 chunks per thread -> deep enough pipeline, wide enough grid.
    long long g = ((long long)n4 + (BLOCK * 8LL) - 1) / (BLOCK * 8LL);
    int G = (int)g;
    if (G < 1) G = 1;
    if (G > 3840) G = 3840;
    // workspace guard: need G*4 (f32) + pad + G*4 (u32)
    size_t cap = (ws_size > 512) ? (ws_size - 256) / 8 : 1;
    if ((size_t)G > cap) G = (int)cap;
    if (G < 1) G = 1;

    float*    lossp = (float*)d_ws;
    unsigned* cntp  = (unsigned*)((char*)d_ws + (((size_t)G * 4 + 255) & ~(size_t)255));

    masked_l1_partial<<<G, BLOCK, 0, stream>>>(outp, lab0, lab1, n, n4, lossp, cntp);
    masked_l1_final<<<1, BLOCK, 0, stream>>>(lossp, cntp, G, (float*)d_out);
}